// ScaledDotProcutionAttention_47871705481473
// MI455X (gfx1250) — compile-verified
//
#include <hip/hip_runtime.h>

typedef float v2f __attribute__((ext_vector_type(2)));
typedef float v8f __attribute__((ext_vector_type(8)));

#define B_ 4
#define H_ 16
#define S_ 2048
#define D_ 64
#define SCALE 0.125f            // 1/sqrt(64)
#define ROWS 16
#define NWAVES 8
#define NTHREADS 256
#define CTILES (S_ / 16)        // 128 column tiles
#define RBLOCKS (S_ / ROWS)     // 128 row blocks per (b,h)

__global__ __launch_bounds__(NTHREADS)
void sdpa_f32_wmma_kernel(const float* __restrict__ q,
                          const float* __restrict__ k,
                          const float* __restrict__ v,
                          const int*   __restrict__ masks,
                          float* __restrict__ out,   // [B,H,S,D]
                          float* __restrict__ mat)   // [B,H,S,S]
{
    // 16-row score stripe, resident across phases: 16*2048*4 = 128 KB
    __shared__ float P[ROWS][S_];
    __shared__ float red[ROWS][16];
    __shared__ float rowmax[ROWS];
    __shared__ float rowsum[ROWS];
    __shared__ float acc_red[4][16][16];   // phase-3 cross-wave reduction

    const int tid  = threadIdx.x;
    const int lane = tid & 31;
    const int wave = tid >> 5;
    const int bh   = blockIdx.x >> 7;          // 0..63  (b*H + h)
    const int rb   = blockIdx.x & (RBLOCKS-1); // 0..127
    const int b    = bh >> 4;
    const int row0 = rb * ROWS;

    const float* qb = q + ((size_t)bh * S_ + row0) * D_;
    const float* kb = k + (size_t)bh * S_ * D_;
    const float* vb = v + (size_t)bh * S_ * D_;
    const int*   mb = masks + (size_t)b * S_ * S_ + (size_t)row0 * S_;

    const int n     = lane & 15;           // N / M index within half-wave
    const int koff  = (lane >> 4) << 1;    // 0 (lanes 0-15) or 2 (lanes 16-31)
    const int mhalf = (lane >> 4) << 3;    // C/D: lanes 16-31 hold M+8

    // ---------------- resident Q A-fragments (16 x 64, fp32) ----------------
    // A 16x4 layout: lanes 0-15 = M rows with K={k0,k0+1}, lanes 16-31 K={k0+2,k0+3}
    v2f aq[16];
    #pragma unroll
    for (int f = 0; f < 16; ++f) {
        const float* p = qb + (size_t)n * D_ + 4 * f + koff;
        aq[f] = *(const v2f*)p;
    }

    // ------- Phase 1: scores = (Q K^T)*scale, mask; 2 tiles/iter for ILP -------
    for (int i = 0; i < CTILES / (2 * NWAVES); ++i) {      // 8 iterations
        const int ct0   = wave + (2 * i) * NWAVES;
        const int ct1   = ct0 + NWAVES;
        const int colb0 = ct0 * 16;
        const int colb1 = ct1 * 16;
        v8f c0 = {}, c1 = {};
        #pragma unroll
        for (int f = 0; f < 16; ++f) {
            // B 4x16 fragment: B[kk][nn] = K[col+nn][k0+kk] (contiguous float2/lane)
            const v2f b0 = *(const v2f*)(kb + (size_t)(colb0 + n) * D_ + 4 * f + koff);
            const v2f b1 = *(const v2f*)(kb + (size_t)(colb1 + n) * D_ + 4 * f + koff);
            c0 = __builtin_amdgcn_wmma_f32_16x16x4_f32(
                     false, aq[f], false, b0, (short)0, c0, false, false);
            c1 = __builtin_amdgcn_wmma_f32_16x16x4_f32(
                     false, aq[f], false, b1, (short)0, c1, false, false);
        }
        #pragma unroll
        for (int e = 0; e < 8; ++e) {
            const int M = e + mhalf;
            float v0 = c0[e] * SCALE;
            float v1 = c1[e] * SCALE;
            if (mb[(size_t)M * S_ + colb0 + n] == 0) v0 = 1e-12f;
            if (mb[(size_t)M * S_ + colb1 + n] == 0) v1 = 1e-12f;
            P[M][colb0 + n] = v0;
            P[M][colb1 + n] = v1;
        }
    }
    __syncthreads();

    // ---------------- Phase 2: exact softmax over the stripe ----------------
    const int r   = tid >> 4;   // row 0..15
    const int j16 = tid & 15;   // 16 threads per row
    float lm = -3.402823466e38f;
    for (int jj = 0; jj < S_ / 16; ++jj)
        lm = fmaxf(lm, P[r][j16 + (jj << 4)]);
    red[r][j16] = lm;
    __syncthreads();
    if (j16 == 0) {
        float m = red[r][0];
        #pragma unroll
        for (int i = 1; i < 16; ++i) m = fmaxf(m, red[r][i]);
        rowmax[r] = m;
    }
    __syncthreads();

    const float rm = rowmax[r];
    float ls = 0.0f;
    for (int jj = 0; jj < S_ / 16; ++jj) {
        const int cidx = j16 + (jj << 4);
        const float e = __expf(P[r][cidx] - rm);
        P[r][cidx] = e;
        ls += e;
    }
    red[r][j16] = ls;
    __syncthreads();
    if (j16 == 0) {
        float s = 0.0f;
        #pragma unroll
        for (int i = 0; i < 16; ++i) s += red[r][i];
        rowsum[r] = s;
    }
    __syncthreads();

    // normalized write of the contiguous 16x2048 stripe (streaming / NT stores:
    // the 1.07 GB matrix output must not evict K/V working set from L2)
    {
        float* mrow = mat + ((size_t)bh * S_ + row0) * S_;
        float* Pf   = &P[0][0];
        for (int idx = tid; idx < ROWS * S_; idx += NTHREADS) {
            const int rr = idx >> 11;           // S_ == 2048
            const float pv = Pf[idx] * (1.0f / rowsum[rr]);
            Pf[idx] = pv;
            __builtin_nontemporal_store(pv, mrow + idx);
        }
    }
    __syncthreads();

    // -------- Phase 3: out = P @ V (16x2048 * 2048x64), dual accum chains --------
    const int dtile = wave & 3;     // which 16-wide D column tile
    const int khalf = wave >> 2;    // split K range in half across wave pairs
    const int dbase = dtile * 16;
    v8f c0 = {}, c1 = {};
    for (int step = 0; step < (S_ / 2) / 8; ++step) {       // 128 iterations
        const int ka = khalf * (S_ / 2) + step * 8 + koff;  // chain 0
        const int kb4 = ka + 4;                             // chain 1
        const v2f a0 = *(const v2f*)&P[n][ka];              // A from LDS (ds_load_b64)
        const v2f a1 = *(const v2f*)&P[n][kb4];
        const v2f b0 = { vb[(size_t)ka  * D_ + dbase + n],
                         vb[(size_t)(ka  + 1) * D_ + dbase + n] };
        const v2f b1 = { vb[(size_t)kb4 * D_ + dbase + n],
                         vb[(size_t)(kb4 + 1) * D_ + dbase + n] };
        c0 = __builtin_amdgcn_wmma_f32_16x16x4_f32(
                 false, a0, false, b0, (short)0, c0, false, false);
        c1 = __builtin_amdgcn_wmma_f32_16x16x4_f32(
                 false, a1, false, b1, (short)0, c1, false, false);
    }
    if (khalf == 1) {
        #pragma unroll
        for (int e = 0; e < 8; ++e)
            acc_red[dtile][e + mhalf][n] = c0[e] + c1[e];
    }
    __syncthreads();
    if (khalf == 0) {
        float* orow = out + ((size_t)bh * S_ + row0) * D_;
        #pragma unroll
        for (int e = 0; e < 8; ++e) {
            const int M = e + mhalf;
            orow[(size_t)M * D_ + dbase + n] = (c0[e] + c1[e]) + acc_red[dtile][M][n];
        }
    }
}

extern "C" void kernel_launch(void* const* d_in, const int* in_sizes, int n_in,
                              void* d_out, int out_size, void* d_ws, size_t ws_size,
                              hipStream_t stream) {
    (void)in_sizes; (void)n_in; (void)d_ws; (void)ws_size; (void)out_size;
    const float* q     = (const float*)d_in[0];
    const float* k     = (const float*)d_in[1];
    const float* v     = (const float*)d_in[2];
    const int*   masks = (const int*)d_in[3];

    float* out = (float*)d_out;                                  // [B,H,S,D] first
    float* mat = out + (size_t)B_ * H_ * S_ * D_;                // then [B,H,S,S]

    dim3 grid(B_ * H_ * RBLOCKS);   // 8192 workgroups
    dim3 block(NTHREADS);           // 8 waves of 32
    sdpa_f32_wmma_kernel<<<grid, block, 0, stream>>>(q, k, v, masks, out, mat);
}